// ICP_1400159339068
// MI455X (gfx1250) — compile-verified
//
#include <hip/hip_runtime.h>

typedef __attribute__((ext_vector_type(2))) float v2f;
typedef __attribute__((ext_vector_type(8))) float v8f;

#define NPTS        2048
#define NTILES      128            // NPTS/16
#define NBATCH      16
#define NSPLIT      16             // blocks per batch for the NN phase
#define PTS_PER_BLK 256
#define NTHREADS_A  256            // 8 wave32
#define NWAVES_A    8
#define ITERS       10

// workspace layout (floats):
//   state:    ws[b*16 + k]   k: 0-8 R, 9-11 t, 12 mse, 13 done, 14-15 pad
//   partials: ws[256 + (b*NSPLIT + s)*17 + k]   (17 sums per split)
#define WS_PART_OFF 256

__device__ __forceinline__ float det3(const float m[3][3]) {
    return m[0][0]*(m[1][1]*m[2][2]-m[1][2]*m[2][1])
         - m[0][1]*(m[1][0]*m[2][2]-m[1][2]*m[2][0])
         + m[0][2]*(m[1][0]*m[2][1]-m[1][1]*m[2][0]);
}

__device__ __forceinline__ void jacobi_rot(float A[3][3], float V[3][3], int p, int q) {
    float apq = A[p][q];
    float scale = fabsf(A[p][p]) + fabsf(A[q][q]);
    if (fabsf(apq) <= 1e-12f * scale + 1e-30f) return;
    float theta = (A[q][q] - A[p][p]) / (2.0f * apq);
    float t = ((theta >= 0.0f) ? 1.0f : -1.0f) / (fabsf(theta) + sqrtf(1.0f + theta*theta));
    float c = 1.0f / sqrtf(1.0f + t*t);
    float s = t * c;
    for (int k = 0; k < 3; ++k) { float akp=A[k][p], akq=A[k][q]; A[k][p]=c*akp - s*akq; A[k][q]=s*akp + c*akq; }
    for (int k = 0; k < 3; ++k) { float apk=A[p][k], aqk=A[q][k]; A[p][k]=c*apk - s*aqk; A[q][k]=s*apk + c*aqk; }
    for (int k = 0; k < 3; ++k) { float vkp=V[k][p], vkq=V[k][q]; V[k][p]=c*vkp - s*vkq; V[k][q]=s*vkp + c*vkq; }
}

// ---------------- init: R = I, t = 0, mse = 0, done = 0 ----------------
__global__ void icp_init(float* __restrict__ ws) {
    int tid = threadIdx.x;
    if (tid < NBATCH * 16) {
        int k = tid & 15;
        float v = (k < 9 && (k % 4) == 0) ? 1.0f : 0.0f;
        ws[tid] = v;
    }
}

// ---------------- phase A: transform + WMMA NN + partial statistics ----------------
__global__ __launch_bounds__(NTHREADS_A, 1)
void icp_nn(const float* __restrict__ src, const float* __restrict__ dst,
            const unsigned char* __restrict__ msk, float* __restrict__ ws)
{
    // packed per-lane B fragments: sB[dt*32 + lane]:
    //   lanes 0-15  : (qx, qy)            -> B VGPR0=K0, VGPR1=K1
    //   lanes 16-31 : (qz, -0.5*dsq|inf)  -> B VGPR0=K2, VGPR1=K3
    __shared__ v2f   sB[NTILES * 32];
    __shared__ float spx[PTS_PER_BLK], spy[PTS_PER_BLK], spz[PTS_PER_BLK], spsq[PTS_PER_BLK];
    __shared__ float sred[NWAVES_A][20];

    const int split = blockIdx.x;
    const int b     = blockIdx.y;
    const int tid   = threadIdx.x;
    const int lane  = tid & 31;
    const int wave  = tid >> 5;
    const bool hi   = (lane >= 16);
    const int l15   = lane & 15;

    float* state = ws + b * 16;
    if (state[13] != 0.0f) return;              // converged: state frozen

    const float* gsx = src + (size_t)b * 3 * NPTS;
    const float* gsy = gsx + NPTS;
    const float* gsz = gsy + NPTS;
    const float* gdx = dst + (size_t)b * 3 * NPTS;
    const float* gdy = gdx + NPTS;
    const float* gdz = gdy + NPTS;
    const unsigned char* gm = msk + (size_t)b * NPTS;

    // build packed B fragments (dest is iteration-invariant but cheap: 26KB from L2)
    for (int n = tid; n < NPTS; n += NTHREADS_A) {
        float x = gdx[n], y = gdy[n], z = gdz[n];
        float dsq = x*x + y*y + z*z;
        float k3  = gm[n] ? (-0.5f * dsq) : (-__builtin_inff());
        int dt = n >> 4, l = n & 15;
        v2f lo; lo.x = x; lo.y = y;
        v2f hh; hh.x = z; hh.y = k3;
        sB[dt*32 + l]      = lo;
        sB[dt*32 + 16 + l] = hh;
    }

    // transform this block's 256 source points: p = R*s + t (one point per thread)
    {
        const float R00=state[0], R01=state[1], R02=state[2];
        const float R10=state[3], R11=state[4], R12=state[5];
        const float R20=state[6], R21=state[7], R22=state[8];
        const float t0=state[9], t1=state[10], t2=state[11];
        int m = split * PTS_PER_BLK + tid;
        float x = gsx[m], y = gsy[m], z = gsz[m];
        float px = R00*x + R01*y + R02*z + t0;
        float py = R10*x + R11*y + R12*z + t1;
        float pz = R20*x + R21*y + R22*z + t2;
        spx[tid] = px; spy[tid] = py; spz[tid] = pz;
        spsq[tid] = px*px + py*py + pz*pz;
    }
    __syncthreads();

    // ---- each wave owns one 16-point src tile; scan all 128 dest tiles ----
    const int ltile = wave;
    const int am = ltile * 16 + l15;
    v2f a;
    a.x = hi ? spz[am] : spx[am];               // A VGPR0 = K0(x) | K2(z)
    a.y = hi ? 1.0f    : spy[am];               // A VGPR1 = K1(y) | K3(1)

    const int rb = ltile * 16 + (hi ? 8 : 0);   // rows owned by this half-wave in D
    float psqr[8];
    #pragma unroll
    for (int r = 0; r < 8; ++r) psqr[r] = spsq[rb + r];

    float md[8]; int mi[8];
    #pragma unroll
    for (int r = 0; r < 8; ++r) { md[r] = __builtin_inff(); mi[r] = l15; }

    const int boff = (hi ? 16 : 0) + l15;
    #pragma unroll 4
    for (int dt = 0; dt < NTILES; ++dt) {
        v2f bb = sB[dt*32 + boff];              // single ds_load_b64 per lane
        v8f c = {0.f,0.f,0.f,0.f,0.f,0.f,0.f,0.f};
        // c[m][n] = p_m . q_n - 0.5*|q_n|^2   (K3 slot folds dsq in)
        c = __builtin_amdgcn_wmma_f32_16x16x4_f32(
                false, a, false, bb, (short)0, c, false, false);
        #pragma unroll
        for (int r = 0; r < 8; ++r) {
            float d2 = fmaf(-2.0f, c[r], psqr[r]);   // |p|^2 + |q|^2 - 2 p.q
            if (d2 < md[r]) { md[r] = d2; mi[r] = dt*16 + l15; }
        }
    }

    // argmin across the 16 lanes of each half (xor keeps halves separate)
    #pragma unroll
    for (int r = 0; r < 8; ++r) {
        #pragma unroll
        for (int off = 8; off >= 1; off >>= 1) {
            float od = __shfl_xor(md[r], off, 32);
            int   oi = __shfl_xor(mi[r], off, 32);
            if (od < md[r] || (od == md[r] && oi < mi[r])) { md[r] = od; mi[r] = oi; }
        }
    }

    // lanes 0 and 16 accumulate the 17 statistics for their 8 rows
    float wsum = 0.f, sumd = 0.f;
    float Ss0=0.f,Ss1=0.f,Ss2=0.f, Sq0=0.f,Sq1=0.f,Sq2=0.f;
    float Hq[9] = {0.f,0.f,0.f,0.f,0.f,0.f,0.f,0.f,0.f};
    if (l15 == 0) {
        const int mg0 = split * PTS_PER_BLK + rb;
        #pragma unroll
        for (int r = 0; r < 8; ++r) {
            int m = mg0 + r;
            float d2 = md[r] < 0.0f ? 0.0f : md[r];     // max(d2, 0)
            sumd += d2;
            float w = (d2 < 9.0f) ? 1.0f : 0.0f;        // sqrt(d2) < 3
            int j = mi[r];
            float qx = gdx[j], qy = gdy[j], qz = gdz[j];
            float sx = gsx[m], sy = gsy[m], sz = gsz[m];
            wsum += w;
            Ss0 += w*sx; Ss1 += w*sy; Ss2 += w*sz;
            Sq0 += w*qx; Sq1 += w*qy; Sq2 += w*qz;
            Hq[0] += w*sx*qx; Hq[1] += w*sx*qy; Hq[2] += w*sx*qz;
            Hq[3] += w*sy*qx; Hq[4] += w*sy*qy; Hq[5] += w*sy*qz;
            Hq[6] += w*sz*qx; Hq[7] += w*sz*qy; Hq[8] += w*sz*qz;
        }
    }

    float vals[17] = {wsum, sumd, Ss0, Ss1, Ss2, Sq0, Sq1, Sq2,
                      Hq[0],Hq[1],Hq[2],Hq[3],Hq[4],Hq[5],Hq[6],Hq[7],Hq[8]};
    #pragma unroll
    for (int k = 0; k < 17; ++k) {
        float v = vals[k];
        #pragma unroll
        for (int off = 16; off >= 1; off >>= 1) v += __shfl_xor(v, off, 32);
        if (lane == 0) sred[wave][k] = v;
    }
    __syncthreads();

    if (tid == 0) {
        float* part = ws + WS_PART_OFF + (size_t)(b * NSPLIT + split) * 17;
        #pragma unroll
        for (int k = 0; k < 17; ++k) {
            float acc = 0.0f;
            for (int w = 0; w < NWAVES_A; ++w) acc += sred[w][k];
            part[k] = acc;        // private slot: deterministic, no atomics
        }
    }
}

// ---------------- phase B: fixed-order reduce + Kabsch/SVD + output ----------------
__global__ void icp_update(float* __restrict__ ws, float* __restrict__ out)
{
    const int b = blockIdx.x;
    if (threadIdx.x != 0) return;
    float* state = ws + b * 16;

    if (state[13] == 0.0f) {
        float S[17];
        #pragma unroll
        for (int k = 0; k < 17; ++k) S[k] = 0.0f;
        for (int s = 0; s < NSPLIT; ++s) {
            const float* p = ws + WS_PART_OFF + (size_t)(b * NSPLIT + s) * 17;
            for (int k = 0; k < 17; ++k) S[k] += p[k];
        }
        float Wsum = S[0];
        float new_mse = S[1] / (float)NPTS;
        float wsd = Wsum + 1e-8f;
        float mus[3] = {S[2]/wsd, S[3]/wsd, S[4]/wsd};
        float muq[3] = {S[5]/wsd, S[6]/wsd, S[7]/wsd};
        float H[3][3];
        for (int i = 0; i < 3; ++i)
            for (int j = 0; j < 3; ++j)
                H[i][j] = S[8 + i*3 + j] - mus[i]*S[5+j] - S[2+i]*muq[j] + Wsum*mus[i]*muq[j];

        // 3x3 SVD via Jacobi eigen of H^T H
        float A[3][3];
        for (int i = 0; i < 3; ++i)
            for (int j = 0; j < 3; ++j)
                A[i][j] = H[0][i]*H[0][j] + H[1][i]*H[1][j] + H[2][i]*H[2][j];
        float V[3][3] = {{1,0,0},{0,1,0},{0,0,1}};
        for (int sweep = 0; sweep < 12; ++sweep) {
            jacobi_rot(A, V, 0, 1);
            jacobi_rot(A, V, 0, 2);
            jacobi_rot(A, V, 1, 2);
        }
        float e[3] = {A[0][0], A[1][1], A[2][2]};
        for (int i = 0; i < 2; ++i)
            for (int j = 0; j < 2 - i; ++j)
                if (e[j] < e[j+1]) {
                    float te = e[j]; e[j] = e[j+1]; e[j+1] = te;
                    for (int k = 0; k < 3; ++k) { float tv = V[k][j]; V[k][j] = V[k][j+1]; V[k][j+1] = tv; }
                }
        float u0[3], u1[3], u2[3];
        for (int i = 0; i < 3; ++i) u0[i] = H[i][0]*V[0][0] + H[i][1]*V[1][0] + H[i][2]*V[2][0];
        float n0 = sqrtf(u0[0]*u0[0] + u0[1]*u0[1] + u0[2]*u0[2]);
        if (n0 > 1e-12f) { u0[0]/=n0; u0[1]/=n0; u0[2]/=n0; } else { u0[0]=1.f; u0[1]=0.f; u0[2]=0.f; }
        for (int i = 0; i < 3; ++i) u1[i] = H[i][0]*V[0][1] + H[i][1]*V[1][1] + H[i][2]*V[2][1];
        float d01 = u1[0]*u0[0] + u1[1]*u0[1] + u1[2]*u0[2];
        for (int i = 0; i < 3; ++i) u1[i] -= d01 * u0[i];
        float n1 = sqrtf(u1[0]*u1[0] + u1[1]*u1[1] + u1[2]*u1[2]);
        if (n1 > 1e-12f) { u1[0]/=n1; u1[1]/=n1; u1[2]/=n1; }
        else {
            float ax0 = (fabsf(u0[0]) < 0.9f) ? 1.f : 0.f;
            float ax1 = 1.f - ax0;
            u1[0] = u0[1]*0.f  - u0[2]*ax1;
            u1[1] = u0[2]*ax0  - u0[0]*0.f;
            u1[2] = u0[0]*ax1  - u0[1]*ax0;
            float nn = sqrtf(u1[0]*u1[0] + u1[1]*u1[1] + u1[2]*u1[2]) + 1e-30f;
            u1[0]/=nn; u1[1]/=nn; u1[2]/=nn;
        }
        u2[0] = u0[1]*u1[2] - u0[2]*u1[1];
        u2[1] = u0[2]*u1[0] - u0[0]*u1[2];
        u2[2] = u0[0]*u1[1] - u0[1]*u1[0];      // det(U) = +1 by construction

        float d = det3(V);
        float Rn[3][3];
        for (int i = 0; i < 3; ++i)
            for (int l = 0; l < 3; ++l)
                Rn[i][l] = V[i][0]*u0[l] + V[i][1]*u1[l] + d*V[i][2]*u2[l];
        float tn[3];
        for (int i = 0; i < 3; ++i)
            tn[i] = muq[i] - (Rn[i][0]*mus[0] + Rn[i][1]*mus[1] + Rn[i][2]*mus[2]);

        for (int i = 0; i < 3; ++i)
            for (int j = 0; j < 3; ++j) state[i*3+j] = Rn[i][j];
        state[9] = tn[0]; state[10] = tn[1]; state[11] = tn[2];
        state[12] = new_mse;
        state[13] = (new_mse < 1e-5f) ? 1.0f : 0.0f;
    }

    // outputs: R (16,3,3) | t (16,3,1) | mse (16,1), concatenated flat
    for (int k = 0; k < 9; ++k) out[b*9 + k] = state[k];
    for (int i = 0; i < 3; ++i) out[16*9 + b*3 + i] = state[9 + i];
    out[16*9 + 16*3 + b] = state[12];
}

extern "C" void kernel_launch(void* const* d_in, const int* in_sizes, int n_in,
                              void* d_out, int out_size, void* d_ws, size_t ws_size,
                              hipStream_t stream) {
    (void)in_sizes; (void)n_in; (void)out_size; (void)ws_size;
    const float* src = (const float*)d_in[0];
    const float* dst = (const float*)d_in[1];
    const unsigned char* msk = (const unsigned char*)d_in[2];
    float* out = (float*)d_out;
    float* ws  = (float*)d_ws;       // needs 256 + 16*16*17 floats (~18.5 KB)

    icp_init<<<dim3(1), dim3(256), 0, stream>>>(ws);
    for (int it = 0; it < ITERS; ++it) {
        icp_nn<<<dim3(NSPLIT, NBATCH), dim3(NTHREADS_A), 0, stream>>>(src, dst, msk, ws);
        icp_update<<<dim3(NBATCH), dim3(32), 0, stream>>>(ws, out);
    }
}